// GCN1_523986010479
// MI455X (gfx1250) — compile-verified
//
#include <hip/hip_runtime.h>
#include <hip/hip_bf16.h>

#define NN 50000
#define NE 600000
#define DF 128

typedef __attribute__((ext_vector_type(2))) float v2f;
typedef __attribute__((ext_vector_type(8))) float v8f;

// ---------------- utility kernels ----------------
__global__ void zero_i32_kernel(int* __restrict__ p, int n) {
    int i = blockIdx.x * blockDim.x + threadIdx.x;
    if (i < n) p[i] = 0;
}
__global__ void zero_f32_kernel(float* __restrict__ p, int n) {
    int i = blockIdx.x * blockDim.x + threadIdx.x;
    if (i < n) p[i] = 0.0f;
}

// ---------------- degree histogram ----------------
__global__ void deg_kernel(const int* __restrict__ src, const int* __restrict__ dst,
                           int* __restrict__ degO, int* __restrict__ degI, int E) {
    int e = blockIdx.x * blockDim.x + threadIdx.x;
    if (e < E) {
        atomicAdd(&degO[src[e]], 1);
        atomicAdd(&degI[dst[e]], 1);
    }
}

// ---------------- single-workgroup chunked exclusive scan over degI ----------------
__global__ void scan_kernel(const int* __restrict__ deg, int* __restrict__ rowoff,
                            int* __restrict__ cursor, int n) {
    __shared__ int sh[1024];
    __shared__ int carry;
    if (threadIdx.x == 0) carry = 0;
    __syncthreads();
    for (int base = 0; base < n; base += 1024) {
        int i = base + (int)threadIdx.x;
        int v = (i < n) ? deg[i] : 0;
        sh[threadIdx.x] = v;
        __syncthreads();
        for (int off = 1; off < 1024; off <<= 1) {
            int t = (threadIdx.x >= (unsigned)off) ? sh[threadIdx.x - off] : 0;
            __syncthreads();
            sh[threadIdx.x] += t;
            __syncthreads();
        }
        int incl = sh[threadIdx.x];
        int excl = incl - v + carry;
        if (i < n) { rowoff[i] = excl; cursor[i] = excl; }
        __syncthreads();
        if (threadIdx.x == 1023) carry += incl;
        __syncthreads();
    }
    if (threadIdx.x == 0) rowoff[n] = carry;
}

// ---------------- counting-sort scatter: edges grouped by dst ----------------
__global__ void scatter_kernel(const int* __restrict__ src, const int* __restrict__ dst,
                               int* __restrict__ cursor, int* __restrict__ csr, int E) {
    int e = blockIdx.x * blockDim.x + threadIdx.x;
    if (e < E) {
        int pos = atomicAdd(&cursor[dst[e]], 1);
        csr[pos] = src[e];
    }
}

// ---------------- deg^{-1/2} ----------------
__global__ void invsqrt_kernel(const int* __restrict__ degO, const int* __restrict__ degI,
                               float* __restrict__ dOinv, float* __restrict__ dIinv, int n) {
    int i = blockIdx.x * blockDim.x + threadIdx.x;
    if (i < n) {
        int a = degO[i]; if (a < 1) a = 1;
        int b = degI[i]; if (b < 1) b = 1;
        dOinv[i] = 1.0f / sqrtf((float)a);
        dIinv[i] = 1.0f / sqrtf((float)b);
    }
}

// ---------------- SpMM: out[r,:] = dIinv[r] * sum_{e in row r} dOinv[src]*x[src,:] ----------------
// one wave per node, float4 per lane; 2-edge unroll for MLP on the index->gather chain
__global__ void spmm_kernel(const float* __restrict__ xin, const int* __restrict__ csr,
                            const int* __restrict__ rowoff, const float* __restrict__ dOinv,
                            const float* __restrict__ dIinv, float* __restrict__ out, int n) {
    int wave = (int)((blockIdx.x * blockDim.x + threadIdx.x) >> 5);
    int lane = threadIdx.x & 31;
    if (wave >= n) return;
    int start = rowoff[wave];
    int end   = rowoff[wave + 1];
    float ax = 0.f, ay = 0.f, az = 0.f, aw = 0.f;
    int e = start;
    for (; e + 1 < end; e += 2) {
        int s0 = csr[e];
        int s1 = csr[e + 1];
        float w0 = dOinv[s0];
        float w1 = dOinv[s1];
        float4 v0 = ((const float4*)(xin + (size_t)s0 * DF))[lane];
        float4 v1 = ((const float4*)(xin + (size_t)s1 * DF))[lane];
        ax += w0 * v0.x + w1 * v1.x;
        ay += w0 * v0.y + w1 * v1.y;
        az += w0 * v0.z + w1 * v1.z;
        aw += w0 * v0.w + w1 * v1.w;
    }
    if (e < end) {
        int s = csr[e];
        float w = dOinv[s];
        float4 v = ((const float4*)(xin + (size_t)s * DF))[lane];
        ax += w * v.x; ay += w * v.y; az += w * v.z; aw += w * v.w;
    }
    float wi = dIinv[wave];
    float4 r; r.x = ax * wi; r.y = ay * wi; r.z = az * wi; r.w = aw * wi;
    ((float4*)(out + (size_t)wave * DF))[lane] = r;
}

// ---------------- GEMM (fp32 WMMA 16x16x4) + bias + ReLU ----------------
// one wave computes a 16x64 strip (4 column tiles): A fragment reused 4x.
// N divides 16 exactly and grid covers [0,N) exactly -> no bounds checks.
__global__ void gemm_relu_kernel(const float* __restrict__ A, const float* __restrict__ W,
                                 const float* __restrict__ bias, float* __restrict__ out) {
    const int lane = threadIdx.x & 31;
    const int half = lane >> 4;   // 0 or 1
    const int l    = lane & 15;
    const int row0 = blockIdx.x * 16;
    const int col0 = blockIdx.y * 64;

    const float* arow  = A + (size_t)(row0 + l) * DF + 2 * half;
    const float* wbase = W + (size_t)(2 * half) * DF + col0 + l;

    v8f c[4] = {};
#pragma unroll
    for (int k0 = 0; k0 < DF; k0 += 4) {
        v2f a = *(const v2f*)(arow + k0);
        const float* wk = wbase + (size_t)k0 * DF;
        v2f b0; b0.x = wk[0];  b0.y = wk[DF + 0];
        v2f b1; b1.x = wk[16]; b1.y = wk[DF + 16];
        v2f b2; b2.x = wk[32]; b2.y = wk[DF + 32];
        v2f b3; b3.x = wk[48]; b3.y = wk[DF + 48];
        c[0] = __builtin_amdgcn_wmma_f32_16x16x4_f32(false, a, false, b0, (short)0, c[0], false, false);
        c[1] = __builtin_amdgcn_wmma_f32_16x16x4_f32(false, a, false, b1, (short)0, c[1], false, false);
        c[2] = __builtin_amdgcn_wmma_f32_16x16x4_f32(false, a, false, b2, (short)0, c[2], false, false);
        c[3] = __builtin_amdgcn_wmma_f32_16x16x4_f32(false, a, false, b3, (short)0, c[3], false, false);
    }
#pragma unroll
    for (int t = 0; t < 4; ++t) {
        int col = col0 + 16 * t + l;
        float bv = bias[col];
#pragma unroll
        for (int v = 0; v < 8; ++v) {
            int row = row0 + v + 8 * half;
            float o = c[t][v] + bv;
            out[(size_t)row * DF + col] = o > 0.0f ? o : 0.0f;
        }
    }
}

// ---------------- GEMM + bias + ReLU, fused global sum reduction ----------------
__global__ void gemm_relu_reduce_kernel(const float* __restrict__ A, const float* __restrict__ W,
                                        const float* __restrict__ bias, float* __restrict__ accum) {
    const int lane = threadIdx.x & 31;
    const int half = lane >> 4;
    const int l    = lane & 15;
    const int row0 = blockIdx.x * 16;
    const int col0 = blockIdx.y * 64;

    const float* arow  = A + (size_t)(row0 + l) * DF + 2 * half;
    const float* wbase = W + (size_t)(2 * half) * DF + col0 + l;

    v8f c[4] = {};
#pragma unroll
    for (int k0 = 0; k0 < DF; k0 += 4) {
        v2f a = *(const v2f*)(arow + k0);
        const float* wk = wbase + (size_t)k0 * DF;
        v2f b0; b0.x = wk[0];  b0.y = wk[DF + 0];
        v2f b1; b1.x = wk[16]; b1.y = wk[DF + 16];
        v2f b2; b2.x = wk[32]; b2.y = wk[DF + 32];
        v2f b3; b3.x = wk[48]; b3.y = wk[DF + 48];
        c[0] = __builtin_amdgcn_wmma_f32_16x16x4_f32(false, a, false, b0, (short)0, c[0], false, false);
        c[1] = __builtin_amdgcn_wmma_f32_16x16x4_f32(false, a, false, b1, (short)0, c[1], false, false);
        c[2] = __builtin_amdgcn_wmma_f32_16x16x4_f32(false, a, false, b2, (short)0, c[2], false, false);
        c[3] = __builtin_amdgcn_wmma_f32_16x16x4_f32(false, a, false, b3, (short)0, c[3], false, false);
    }
    float s = 0.0f;
#pragma unroll
    for (int t = 0; t < 4; ++t) {
        float bv = bias[col0 + 16 * t + l];
#pragma unroll
        for (int v = 0; v < 8; ++v) {
            float o = c[t][v] + bv;
            s += (o > 0.0f ? o : 0.0f);
        }
    }
    for (int off = 16; off >= 1; off >>= 1) s += __shfl_down(s, off, 32);
    if (lane == 0) atomicAdd(accum, s);
}

__global__ void finalize_kernel(const float* __restrict__ accum, float* __restrict__ out) {
    if (threadIdx.x == 0 && blockIdx.x == 0)
        out[0] = accum[0] * (1.0f / (4.0f * (float)NN * (float)DF));
}

// ---------------- host launch ----------------
extern "C" void kernel_launch(void* const* d_in, const int* in_sizes, int n_in,
                              void* d_out, int out_size, void* d_ws, size_t ws_size,
                              hipStream_t stream) {
    (void)in_sizes; (void)n_in; (void)out_size; (void)ws_size;
    char* p = (char*)d_ws;
    auto carve = [&](size_t bytes) -> void* {
        void* r = (void*)p;
        p += (bytes + 255) & ~(size_t)255;
        return r;
    };
    float* accum  = (float*)carve(sizeof(float));
    int*   degO   = (int*)carve(NN * sizeof(int));
    int*   degI   = (int*)carve(NN * sizeof(int));
    int*   rowoff = (int*)carve((NN + 1) * sizeof(int));
    int*   cursor = (int*)carve(NN * sizeof(int));
    int*   csr    = (int*)carve(NE * sizeof(int));
    float* dOinv  = (float*)carve(NN * sizeof(float));
    float* dIinv  = (float*)carve(NN * sizeof(float));
    float* bufA   = (float*)carve((size_t)NN * DF * sizeof(float));
    float* bufB   = (float*)carve((size_t)NN * DF * sizeof(float));

    const float* W1 = (const float*)d_in[12];
    const float* b1 = (const float*)d_in[13];
    const float* W2 = (const float*)d_in[14];
    const float* b2 = (const float*)d_in[15];

    zero_f32_kernel<<<1, 32, 0, stream>>>(accum, 1);

    dim3 gGemm(NN / 16, DF / 64);      // 3125 x 2, one wave per 16x64 strip
    int  bEdge = 256, gEdge = (NE + bEdge - 1) / bEdge;
    int  bNode = 256, gNode = (NN + bNode - 1) / bNode;
    int  gSpmm = (NN * 32 + 255) / 256;

    for (int g = 0; g < 4; ++g) {
        const float* x   = (const float*)d_in[3 * g + 0];
        const int*   src = (const int*)d_in[3 * g + 1];
        const int*   dst = (const int*)d_in[3 * g + 2];

        zero_i32_kernel<<<gNode, bNode, 0, stream>>>(degO, NN);
        zero_i32_kernel<<<gNode, bNode, 0, stream>>>(degI, NN);
        deg_kernel<<<gEdge, bEdge, 0, stream>>>(src, dst, degO, degI, NE);
        scan_kernel<<<1, 1024, 0, stream>>>(degI, rowoff, cursor, NN);
        scatter_kernel<<<gEdge, bEdge, 0, stream>>>(src, dst, cursor, csr, NE);
        invsqrt_kernel<<<gNode, bNode, 0, stream>>>(degO, degI, dOinv, dIinv, NN);

        spmm_kernel<<<gSpmm, 256, 0, stream>>>(x, csr, rowoff, dOinv, dIinv, bufA, NN);
        gemm_relu_kernel<<<gGemm, 32, 0, stream>>>(bufA, W1, b1, bufB);

        spmm_kernel<<<gSpmm, 256, 0, stream>>>(bufB, csr, rowoff, dOinv, dIinv, bufA, NN);
        gemm_relu_reduce_kernel<<<gGemm, 32, 0, stream>>>(bufA, W2, b2, accum);
    }

    finalize_kernel<<<1, 32, 0, stream>>>(accum, (float*)d_out);
}